// LossCompute_25228637896745
// MI455X (gfx1250) — compile-verified
//
#include <hip/hip_runtime.h>
#include <hip/hip_bf16.h>

typedef __attribute__((ext_vector_type(2))) float v2f;
typedef __attribute__((ext_vector_type(8))) float v8f;
typedef __attribute__((ext_vector_type(4))) int   v4i;

#define NUM_GRAPHS 1024
#define CLAUSES_PER_WAVE 8192

__device__ __forceinline__ void atomAddF(float* p, float v) {
#if defined(__HIP_DEVICE_COMPILE__) && defined(__AMDGCN__)
    unsafeAtomicAdd(p, v);   // hardware global_atomic_add_f32
#else
    atomicAdd(p, v);
#endif
}

// ---------------- zero workspace ----------------
__global__ void zero_kernel(float* __restrict__ p, long n) {
    long t = (long)blockIdx.x * blockDim.x + threadIdx.x;
    long stride = (long)gridDim.x * blockDim.x;
    long n4 = n >> 2;
    float4 z = make_float4(0.f, 0.f, 0.f, 0.f);
    for (long i = t; i < n4; i += stride) ((float4*)p)[i] = z;
    for (long i = (n4 << 2) + t; i < n; i += stride) p[i] = 0.f;
}

// ---------------- edge scatter: num/den atomics ----------------
template <int NEG>
__global__ void edge_kernel(const int* __restrict__ cls, const int* __restrict__ var,
                            const float* __restrict__ xv,
                            float* __restrict__ num, float* __restrict__ den, long E) {
    long t = (long)blockIdx.x * blockDim.x + threadIdx.x;
    long stride = (long)gridDim.x * blockDim.x;
    long n4 = E >> 2;
    for (long i = t; i < n4; i += stride) {
        long e = i << 2;
        // adjacency is streamed once: non-temporal so it does not evict the
        // L2-resident clause accumulators (67MB) and xv (8MB) from the 192MB L2
        v4i c4 = __builtin_nontemporal_load((const v4i*)(cls + e));
        v4i v4 = __builtin_nontemporal_load((const v4i*)(var + e));
#pragma unroll
        for (int k = 0; k < 4; ++k) {
            int c = c4[k];
            float x = xv[v4[k]];
            if (NEG) x = 1.0f - x;
            float w = __expf(10.0f * x);
            atomAddF(num + c, x * w);
            atomAddF(den + c, w);
        }
    }
    for (long e = (n4 << 2) + t; e < E; e += stride) {
        int c = cls[e]; int v = var[e];
        float x = xv[v];
        if (NEG) x = 1.0f - x;
        float w = __expf(10.0f * x);
        atomAddF(num + c, x * w);
        atomAddF(den + c, w);
    }
}

// ---------------- WMMA row-sum helpers ----------------
// After accumulating D = A*ones + C repeatedly, every column j of D holds the
// accumulated row sums.  Column N=0 lives in lane 0 (M=0..7, vgpr 0..7) and
// lane 16 (M=8..15).  Total = shfl(lane-sum, 0) + shfl(lane-sum, 16).
__device__ __forceinline__ float wmma_tile_total(const v8f& c) {
    float s = c[0] + c[1] + c[2] + c[3] + c[4] + c[5] + c[6] + c[7];
    return __shfl(s, 0) + __shfl(s, 16);
}

__device__ __forceinline__ void flush_sm(v8f& csm, int& cnt, int& cur_g,
                                         float* gsum, float* gcnt, int lane) {
    if (cur_g >= 0 && cnt > 0) {           // wave-uniform condition
        float tot = wmma_tile_total(csm);
        if (lane == 0) {
            atomAddF(gsum + cur_g, tot);
            atomAddF(gcnt + cur_g, (float)cnt);
        }
    }
    csm = (v8f){};
    cnt = 0;
}

// ---------------- clause -> graph segmented reduction via WMMA ----------------
__global__ void clause_kernel(const float* __restrict__ num, const float* __restrict__ den,
                              const int* __restrict__ gr,
                              float* __restrict__ gsum, float* __restrict__ gcnt,
                              float* __restrict__ relu_acc, int NC) {
    const int lane = threadIdx.x & 31;
    const long wave = ((long)blockIdx.x * blockDim.x + threadIdx.x) >> 5;
    const long wstart = wave * (long)CLAUSES_PER_WAVE;
    long wend = wstart + CLAUSES_PER_WAVE;
    if (wend > (long)NC) wend = (long)NC;

    v8f csm = (v8f){};   // per-graph sum of sm
    v8f crl = (v8f){};   // graph-agnostic sum of relu(10*(sm-0.45))
    const v2f vone = {1.0f, 1.0f};
    int cur_g = -1;
    int cnt = 0;

    for (long base = wstart; base < wend; base += 64) {
        if (base + 64 <= wend) {                       // full 64-clause chunk (uniform)
            int gf = 0, gl = 0;
            if (lane == 0) { gf = gr[base]; gl = gr[base + 63]; }
            gf = __shfl(gf, 0); gl = __shfl(gl, 0);    // uniform broadcast

            // A-matrix 16x4 f32 layout: lane = m + 16*(k>=2); vgpr0/1 = k pair
            long idx = base + (long)(((lane & 15) << 2) + ((lane >> 4) << 1));
            float2 n2 = *(const float2*)(num + idx);
            float2 d2 = *(const float2*)(den + idx);
            v2f a; a.x = n2.x / d2.x; a.y = n2.y / d2.y;
            v2f r; r.x = fmaxf(0.0f, 10.0f * (a.x - 0.45f));
                   r.y = fmaxf(0.0f, 10.0f * (a.y - 0.45f));

            // relu penalty: always WMMA-accumulate (graph independent)
            crl = __builtin_amdgcn_wmma_f32_16x16x4_f32(
                false, r, false, vone, (short)0, crl, false, false);

            if (gf == gl) {                            // chunk within one graph
                if (gf != cur_g) {
                    flush_sm(csm, cnt, cur_g, gsum, gcnt, lane);
                    cur_g = gf;
                }
                csm = __builtin_amdgcn_wmma_f32_16x16x4_f32(
                    false, a, false, vone, (short)0, csm, false, false);
                cnt += 64;
            } else {                                   // graph boundary inside chunk: scalar
                flush_sm(csm, cnt, cur_g, gsum, gcnt, lane);
                cur_g = -1;
                int g0 = gr[idx], g1 = gr[idx + 1];
                atomAddF(gsum + g0, a.x); atomAddF(gcnt + g0, 1.0f);
                atomAddF(gsum + g1, a.y); atomAddF(gcnt + g1, 1.0f);
            }
        } else {                                       // tail (<64): scalar, uniform entry
            for (long j = base + lane; j < wend; j += 32) {
                float s = num[j] / den[j];
                int g = gr[j];
                atomAddF(gsum + g, s);
                atomAddF(gcnt + g, 1.0f);
                float rr = fmaxf(0.0f, 10.0f * (s - 0.45f));
                atomAddF(relu_acc, rr);
            }
        }
    }
    flush_sm(csm, cnt, cur_g, gsum, gcnt, lane);
    float rtot = wmma_tile_total(crl);
    if (lane == 0) atomAddF(relu_acc, rtot);
}

// ---------------- finalize: loss scalar ----------------
__global__ void final_kernel(const float* __restrict__ gsum, const float* __restrict__ gcnt,
                             const float* __restrict__ relu_acc,
                             const int* __restrict__ is_train, float* __restrict__ out) {
    __shared__ float red[NUM_GRAPHS];
    int g = threadIdx.x;
    float d = gsum[g] / gcnt[g] - 1.0f;
    red[g] = d * d;
    __syncthreads();
    for (int s = NUM_GRAPHS / 2; s > 0; s >>= 1) {
        if (g < s) red[g] += red[g + s];
        __syncthreads();
    }
    if (g == 0) {
        float loss = red[0] / (float)NUM_GRAPHS;
        if (*is_train) loss -= relu_acc[0] * 0.005f;
        out[0] = loss;
    }
}

extern "C" void kernel_launch(void* const* d_in, const int* in_sizes, int n_in,
                              void* d_out, int out_size, void* d_ws, size_t ws_size,
                              hipStream_t stream) {
    const float* xv      = (const float*)d_in[0];
    const int*   adj_pos = (const int*)d_in[1];
    const int*   adj_neg = (const int*)d_in[2];
    const int*   gr      = (const int*)d_in[4];
    const int*   is_train= (const int*)d_in[5];

    const long E  = (long)in_sizes[1] / 2;   // edges per polarity
    const long NC = (long)in_sizes[4];       // number of clauses

    // workspace layout: num[NC] | den[NC] | gsum[1024] | gcnt[1024] | relu[1]
    float* num  = (float*)d_ws;
    float* den  = num + NC;
    float* gsum = den + NC;
    float* gcnt = gsum + NUM_GRAPHS;
    float* relu = gcnt + NUM_GRAPHS;
    const long Z = 2 * NC + 2 * NUM_GRAPHS + 1;

    {
        long z4 = (Z + 3) >> 2;
        int blocks = (int)((z4 + 255) / 256);
        if (blocks > 32768) blocks = 32768;
        zero_kernel<<<blocks, 256, 0, stream>>>((float*)d_ws, Z);
    }
    {
        long n4 = (E + 3) >> 2;
        int blocks = (int)((n4 + 255) / 256);
        if (blocks > 32768) blocks = 32768;
        edge_kernel<0><<<blocks, 256, 0, stream>>>(adj_pos, adj_pos + E, xv, num, den, E);
        edge_kernel<1><<<blocks, 256, 0, stream>>>(adj_neg, adj_neg + E, xv, num, den, E);
    }
    {
        long waves = (NC + CLAUSES_PER_WAVE - 1) / CLAUSES_PER_WAVE;
        int blocks = (int)((waves * 32 + 255) / 256);
        clause_kernel<<<blocks, 256, 0, stream>>>(num, den, gr, gsum, gcnt, relu, (int)NC);
    }
    final_kernel<<<1, NUM_GRAPHS, 0, stream>>>(gsum, gcnt, relu, is_train, (float*)d_out);
}